// get_model_78202764525640
// MI455X (gfx1250) — compile-verified
//
#include <hip/hip_runtime.h>

typedef float v2f __attribute__((ext_vector_type(2)));
typedef float v8f __attribute__((ext_vector_type(8)));

#define BB 4
#define NN 8192
#define CC 3
#define DD 128
#define WAVES_PER_BLOCK 8
#define XTILES 4   // X tiles (16 rows each) register-blocked per wave

// ---------------------------------------------------------------------------
// Kernel 1: mean_e[b][d] = mean_n relu(enc[b,n,:] @ W1[:,d] + b1[d])
// Tiny: 4 blocks x 128 threads, fully deterministic.
// ---------------------------------------------------------------------------
__global__ void mean_feat_kernel(const float* __restrict__ enc,
                                 const float* __restrict__ W1,
                                 const float* __restrict__ b1,
                                 float* __restrict__ mean_e) {
    const int b = blockIdx.x;
    const int d = threadIdx.x;            // 0..127
    const float w0 = W1[0 * DD + d];
    const float w1 = W1[1 * DD + d];
    const float w2 = W1[2 * DD + d];
    const float bc = b1[d];
    const float* p = enc + (size_t)b * NN * CC;
    float s = 0.0f;
    for (int n = 0; n < NN; ++n) {
        float h = fmaf(p[n * 3 + 0], w0,
                  fmaf(p[n * 3 + 1], w1,
                  fmaf(p[n * 3 + 2], w2, bc)));
        s += fmaxf(h, 0.0f);
    }
    mean_e[b * DD + d] = s * (1.0f / (float)NN);
}

// ---------------------------------------------------------------------------
// Kernel 2: warped[i] = decoder[i] + b2 + feat @ W2, with
//   f_d = relu(dec @ W1 + b1), feat = [f_d + mean_e, f_d]
// One thread per point (B*N = 32768).
// ---------------------------------------------------------------------------
__global__ void warp_kernel(const float* __restrict__ dec,
                            const float* __restrict__ W1,
                            const float* __restrict__ b1,
                            const float* __restrict__ W2,
                            const float* __restrict__ b2,
                            const float* __restrict__ mean_e,
                            float* __restrict__ warped) {
    const int i = blockIdx.x * blockDim.x + threadIdx.x;
    if (i >= BB * NN) return;
    const int b = i / NN;
    const float p0 = dec[i * 3 + 0];
    const float p1 = dec[i * 3 + 1];
    const float p2 = dec[i * 3 + 2];
    const float* me = mean_e + b * DD;
    float a0 = 0.0f, a1 = 0.0f, a2 = 0.0f;
    for (int d = 0; d < DD; ++d) {
        float h = fmaf(p0, W1[0 * DD + d],
                  fmaf(p1, W1[1 * DD + d],
                  fmaf(p2, W1[2 * DD + d], b1[d])));
        h = fmaxf(h, 0.0f);
        const float cr = h + me[d];
        a0 = fmaf(cr, W2[d * 3 + 0], fmaf(h, W2[(DD + d) * 3 + 0], a0));
        a1 = fmaf(cr, W2[d * 3 + 1], fmaf(h, W2[(DD + d) * 3 + 1], a1));
        a2 = fmaf(cr, W2[d * 3 + 2], fmaf(h, W2[(DD + d) * 3 + 2], a2));
    }
    warped[i * 3 + 0] = a0 + b2[0] + p0;
    warped[i * 3 + 1] = a1 + b2[1] + p1;
    warped[i * 3 + 2] = a2 + b2[2] + p2;
}

// ---------------------------------------------------------------------------
// Kernel 3: WMMA chamfer row-min, register-blocked 4 X-tiles per wave,
// y-sweep unrolled by 2 so the min reduction fuses to v_min3_num_f32.
// Each wave owns 64 X rows (4 tiles) of one batch and sweeps all 512 Y tiles.
//   A row m = [x0,x1,x2,1], B col n = [-2y0,-2y1,-2y2,|y|^2], C = |x_m|^2
//   => D[m][n] = |x|^2 - 2 x.y + |y|^2 = squared distance, straight from the
//      matrix unit (one v_wmma_f32_16x16x4_f32 per 16x16 tile).
// Writes per-wave partial sum (already scaled) to `partials[pidx]`.
// ---------------------------------------------------------------------------
__global__ void chamfer_rowmin_kernel(const float* __restrict__ X,
                                      const float* __restrict__ Y,
                                      float* __restrict__ partials,
                                      int partial_base,
                                      float scale) {
    const int wave = threadIdx.x >> 5;
    const int lane = threadIdx.x & 31;
    const int b = blockIdx.y;
    const int xg = blockIdx.x * WAVES_PER_BLOCK + wave;   // x-tile group
    const float* Xb = X + (size_t)b * NN * 3;
    const float* Yb = Y + (size_t)b * NN * 3;

    const int m = lane & 15;            // row / col within a 16-tile
    const int hi = lane >> 4;           // 0: K={0,1}  1: K={2,3}

    // ---- A matrices + C inits for the 4 X tiles (all loop-invariant) ----
    v2f a[XTILES];
    v8f cinit[XTILES];
    float rmin[XTILES][8];
#pragma unroll
    for (int t = 0; t < XTILES; ++t) {
        const int x0 = (xg * XTILES + t) * 16;
        const float px0 = Xb[(x0 + m) * 3 + 0];
        const float px1 = Xb[(x0 + m) * 3 + 1];
        const float px2 = Xb[(x0 + m) * 3 + 2];
        const float myx2 = fmaf(px0, px0, fmaf(px1, px1, px2 * px2));
        a[t].x = hi ? px2 : px0;
        a[t].y = hi ? 1.0f : px1;
#pragma unroll
        for (int r = 0; r < 8; ++r) {
            cinit[t][r] = __shfl(myx2, r + 8 * hi, 32);
            rmin[t][r] = 3.402823466e+38f;
        }
    }

    // ---- sweep all Y tiles, 2 per iteration: each B-build feeds XTILES
    //      WMMAs; the two results fold into the accumulator via min3 ----
    for (int yt = 0; yt < NN / 16; yt += 2) {
        const float* q0p = Yb + (size_t)(yt * 16 + m) * 3;
        const float q00 = q0p[0];
        const float q01 = q0p[1];
        const float q02 = q0p[2];
        const float* q1p = q0p + 48;     // next 16-point tile
        const float q10 = q1p[0];
        const float q11 = q1p[1];
        const float q12 = q1p[2];

        v2f bv0, bv1;
        bv0.x = hi ? (-2.0f * q02) : (-2.0f * q00);
        bv0.y = hi ? fmaf(q00, q00, fmaf(q01, q01, q02 * q02)) : (-2.0f * q01);
        bv1.x = hi ? (-2.0f * q12) : (-2.0f * q10);
        bv1.y = hi ? fmaf(q10, q10, fmaf(q11, q11, q12 * q12)) : (-2.0f * q11);

#pragma unroll
        for (int t = 0; t < XTILES; ++t) {
            v8f d0 = __builtin_amdgcn_wmma_f32_16x16x4_f32(
                false, a[t], false, bv0, (short)0, cinit[t], false, false);
            v8f d1 = __builtin_amdgcn_wmma_f32_16x16x4_f32(
                false, a[t], false, bv1, (short)0, cinit[t], false, false);
#pragma unroll
            for (int r = 0; r < 8; ++r)
                rmin[t][r] = fminf(rmin[t][r], fminf((float)d0[r], (float)d1[r]));
        }
    }

    // ---- reduce mins across the 16 columns (lanes within each half) ----
    float s = 0.0f;
#pragma unroll
    for (int t = 0; t < XTILES; ++t) {
#pragma unroll
        for (int r = 0; r < 8; ++r) {
            float v = rmin[t][r];
            v = fminf(v, __shfl_xor(v, 1, 32));
            v = fminf(v, __shfl_xor(v, 2, 32));
            v = fminf(v, __shfl_xor(v, 4, 32));
            v = fminf(v, __shfl_xor(v, 8, 32));
            s += v;                      // rows 0..7 (lo half) / 8..15 (hi half)
        }
    }
    s += __shfl_xor(s, 16, 32);          // combine the two halves

    if (lane == 0) {
        const int pidx = partial_base + (b * gridDim.x + blockIdx.x) * WAVES_PER_BLOCK + wave;
        partials[pidx] = s * scale;
    }
}

// ---------------------------------------------------------------------------
// Kernel 4: deterministic tree reduction of all partials -> loss scalar.
// ---------------------------------------------------------------------------
__global__ void reduce_loss_kernel(const float* __restrict__ partials, int n,
                                   float* __restrict__ loss) {
    __shared__ float sm[256];
    float s = 0.0f;
    for (int i = threadIdx.x; i < n; i += 256) s += partials[i];
    sm[threadIdx.x] = s;
    __syncthreads();
    for (int off = 128; off > 0; off >>= 1) {
        if (threadIdx.x < off) sm[threadIdx.x] += sm[threadIdx.x + off];
        __syncthreads();
    }
    if (threadIdx.x == 0) *loss = sm[0];
}

// ---------------------------------------------------------------------------
extern "C" void kernel_launch(void* const* d_in, const int* in_sizes, int n_in,
                              void* d_out, int out_size, void* d_ws, size_t ws_size,
                              hipStream_t stream) {
    const float* enc = (const float*)d_in[0];   // [B,N,3]
    const float* dec = (const float*)d_in[1];   // [B,N,3]
    const float* W1  = (const float*)d_in[2];   // [3,128]
    const float* b1  = (const float*)d_in[3];   // [128]
    const float* W2  = (const float*)d_in[4];   // [256,3]
    const float* b2  = (const float*)d_in[5];   // [3]

    float* out    = (float*)d_out;
    float* warped = out;                        // B*N*3 floats
    float* loss   = out + (size_t)BB * NN * 3;  // 1 float

    float* mean_e   = (float*)d_ws;             // B*D floats
    float* partials = mean_e + BB * DD;         // 2 * 512 floats

    // MLP: channel means, then per-point warp.
    mean_feat_kernel<<<BB, DD, 0, stream>>>(enc, W1, b1, mean_e);
    warp_kernel<<<(BB * NN + 255) / 256, 256, 0, stream>>>(dec, W1, b1, W2, b2,
                                                           mean_e, warped);

    // Chamfer: both directions via the same WMMA row-min kernel.
    const int groups = NN / 16 / XTILES;                       // 128 x-tile groups
    dim3 cgrid(groups / WAVES_PER_BLOCK, BB);                  // 16 x 4
    const int waves_per_pass = groups * BB;                    // 512
    const float scale = 1.0f / (float)(BB * NN);
    chamfer_rowmin_kernel<<<cgrid, 32 * WAVES_PER_BLOCK, 0, stream>>>(
        enc, warped, partials, 0, scale);
    chamfer_rowmin_kernel<<<cgrid, 32 * WAVES_PER_BLOCK, 0, stream>>>(
        warped, enc, partials, waves_per_pass, scale);

    // Deterministic final sum -> loss.
    reduce_loss_kernel<<<1, 256, 0, stream>>>(partials, 2 * waves_per_pass, loss);
}